// EstimateAdj_69836168233271
// MI455X (gfx1250) — compile-verified
//
#include <hip/hip_runtime.h>

typedef __attribute__((ext_vector_type(2))) float v2f;
typedef __attribute__((ext_vector_type(8))) float v8f;

#define FDIM 128

// ---------------------------------------------------------------------------
// deg / dinv / accumulator setup
// ---------------------------------------------------------------------------
__global__ void __launch_bounds__(256) zero_init_kernel(unsigned* __restrict__ deg,
                                                        float* __restrict__ accum,
                                                        unsigned* __restrict__ cnt, int n) {
    int i = blockIdx.x * blockDim.x + threadIdx.x;
    if (i < n) deg[i] = 0u;
    if (i < 2) { accum[i] = 0.0f; cnt[i] = 0u; }
}

__global__ void __launch_bounds__(256) count_deg_kernel(const int* __restrict__ dst,
                                                        unsigned* __restrict__ deg, int E) {
    int e = blockIdx.x * blockDim.x + threadIdx.x;
    if (e < E) atomicAdd(&deg[dst[e]], 1u);
}

__global__ void __launch_bounds__(256) dinv_kernel(const unsigned* __restrict__ deg,
                                                   float* __restrict__ dinv, int n) {
    int i = blockIdx.x * blockDim.x + threadIdx.x;
    if (i < n) dinv[i] = rsqrtf((float)deg[i] + 1.0f);  // +1 = self loop
}

// ---------------------------------------------------------------------------
// GEMM + fused self-loop/bias epilogue:
//   Hout[n][j]   = sum_k act(A[n][k]) * W[k][j]
//   AggOut[n][j] = Hout[n][j] * dinv[n]^2 + bias[j]
// One wave per 16-row strip; 8 v8f accumulators cover all 128 output columns.
// V_WMMA_F32_16X16X4_F32 (exact f32). W staged in LDS, K-pair interleaved so
// each B fragment is a single contiguous ds_load_b64 (no operand repacking).
// ---------------------------------------------------------------------------
template <bool RELU>
__global__ void __launch_bounds__(256) gemm_wmma_kernel(const float* __restrict__ A,
                                                        const float* __restrict__ W,
                                                        float* __restrict__ Hout,
                                                        float* __restrict__ AggOut,
                                                        const float* __restrict__ dinv,
                                                        const float* __restrict__ bias,
                                                        int nrows) {
    // wlds[(k>>1)*256 + n*2 + (k&1)] = W[k][n]  -> B frag = contiguous pair
    __shared__ float wlds[FDIM * FDIM];
    for (int i = threadIdx.x; i < FDIM * FDIM; i += 256) {
        int k = i >> 7, n = i & 127;
        wlds[((k >> 1) << 8) + (n << 1) + (k & 1)] = W[i];
    }
    __syncthreads();

    const int lane  = threadIdx.x & 31;
    const int wave  = threadIdx.x >> 5;
    const int strip = blockIdx.x * 8 + wave;
    const int nstrips = nrows >> 4;
    if (strip >= nstrips) return;  // wave-uniform: EXEC stays all-ones for WMMA

    const int row0  = strip << 4;
    const int m     = lane & 15;          // A row / B col within tile
    const int khalf = lane >> 4;          // 0: K={0,1}, 1: K={2,3} of each step
    const int kbase = khalf << 1;

    const float* arow = A + (size_t)(row0 + m) * FDIM;

    v8f acc[8];
#pragma unroll
    for (int t = 0; t < 8; ++t) acc[t] = (v8f)(0.0f);

    for (int k0 = 0; k0 < FDIM; k0 += 4) {
        v2f a = *(const v2f*)(arow + k0 + kbase);
        if (RELU) { a.x = fmaxf(a.x, 0.0f); a.y = fmaxf(a.y, 0.0f); }
        const int bbase = ((k0 >> 1) + khalf) * 256 + (m << 1);
#pragma unroll
        for (int t = 0; t < 8; ++t) {
            v2f b = *(const v2f*)(&wlds[bbase + t * 32]);
            acc[t] = __builtin_amdgcn_wmma_f32_16x16x4_f32(
                /*neg_a=*/false, a, /*neg_b=*/false, b,
                /*c_mod=*/(short)0, acc[t], /*reuse_a=*/false, /*reuse_b=*/false);
        }
    }

    const int rbase = row0 + (khalf << 3);  // lanes 16-31 hold rows +8
    float dr[8];
#pragma unroll
    for (int r = 0; r < 8; ++r) { float t = dinv[rbase + r]; dr[r] = t * t; }
#pragma unroll
    for (int t = 0; t < 8; ++t) {
        float bt = bias[t * 16 + m];
#pragma unroll
        for (int r = 0; r < 8; ++r) {
            float hv = acc[t][r];
            size_t off = (size_t)(rbase + r) * FDIM + t * 16 + m;
            Hout[off]   = hv;
            AggOut[off] = hv * dr[r] + bt;   // self-loop + bias prefill
        }
    }
}

// ---------------------------------------------------------------------------
// out[dst][:] += h[src][:] * dinv[src]*dinv[dst]   -- one wave per edge,
// lanes stride the 128 features (float4/lane), f32 global atomics.
// ---------------------------------------------------------------------------
__global__ void __launch_bounds__(256) scatter_kernel(const float4* __restrict__ h,
                                                      float* __restrict__ out,
                                                      const int* __restrict__ src,
                                                      const int* __restrict__ dst,
                                                      const float* __restrict__ dinv, int E) {
    int lane = threadIdx.x & 31;
    int e = blockIdx.x * (blockDim.x >> 5) + (threadIdx.x >> 5);
    if (e >= E) return;
    int s = src[e], d = dst[e];
    float nrm = dinv[s] * dinv[d];
    float4 v = h[(size_t)s * 32 + lane];
    float* o = out + (size_t)d * FDIM + lane * 4;
    atomicAdd(o + 0, v.x * nrm);
    atomicAdd(o + 1, v.y * nrm);
    atomicAdd(o + 2, v.z * nrm);
    atomicAdd(o + 3, v.w * nrm);
}

// ---------------------------------------------------------------------------
// masked edge similarity loss: for edges with src<dst,
//   loss += (dot(rep[src],rep[dst]) - target)^2 ; cnt++
// grid-stride wave-per-edge; shuffle reduction; one atomic per wave at end.
// ---------------------------------------------------------------------------
__global__ void __launch_bounds__(256) edge_loss_kernel(const float4* __restrict__ rep,
                                                        const int* __restrict__ src,
                                                        const int* __restrict__ dst, int E,
                                                        float target,
                                                        float* __restrict__ loss_acc,
                                                        unsigned* __restrict__ cnt_acc) {
    int lane   = threadIdx.x & 31;
    int wave   = blockIdx.x * (blockDim.x >> 5) + (threadIdx.x >> 5);
    int nwaves = gridDim.x * (blockDim.x >> 5);
    float local = 0.0f;
    unsigned cnt = 0;
    for (int e = wave; e < E; e += nwaves) {
        int s = src[e], d = dst[e];
        if (s < d) {  // wave-uniform branch
            float4 a = rep[(size_t)s * 32 + lane];
            float4 b = rep[(size_t)d * 32 + lane];
            float p = a.x * b.x + a.y * b.y + a.z * b.z + a.w * b.w;
#pragma unroll
            for (int off = 16; off > 0; off >>= 1) p += __shfl_down(p, off, 32);
            if (lane == 0) {
                float dl = p - target;
                local += dl * dl;
                cnt++;
            }
        }
    }
    if (lane == 0 && cnt > 0) {
        atomicAdd(loss_acc, local);
        atomicAdd(cnt_acc, cnt);
    }
}

__global__ void finalize_kernel(const float* __restrict__ accum,
                                const unsigned* __restrict__ cnt,
                                float* __restrict__ out_scalar, float n_nodes) {
    if (threadIdx.x == 0 && blockIdx.x == 0) {
        float tot = accum[0] + accum[1];
        float c = (float)(cnt[0] + cnt[1]);
        *out_scalar = tot * n_nodes / c;
    }
}

// ---------------------------------------------------------------------------
extern "C" void kernel_launch(void* const* d_in, const int* in_sizes, int n_in,
                              void* d_out, int out_size, void* d_ws, size_t ws_size,
                              hipStream_t stream) {
    const float* features = (const float*)d_in[0];
    const int*   ei       = (const int*)d_in[1];
    const int*   nei      = (const int*)d_in[2];
    const float* w1       = (const float*)d_in[3];
    const float* b1       = (const float*)d_in[4];
    const float* w2       = (const float*)d_in[5];
    const float* b2       = (const float*)d_in[6];

    const int N  = in_sizes[0] / FDIM;  // 100000
    const int E  = in_sizes[1] / 2;     // 1.6M
    const int En = in_sizes[2] / 2;     // 0.5M
    const int* src  = ei;
    const int* dst  = ei + E;
    const int* nsrc = nei;
    const int* ndst = nei + En;

    // workspace layout
    char* ws = (char*)d_ws;
    float*    bufA = (float*)ws;                                 // N*128 f32 (h1 / h2)
    float*    bufB = (float*)(ws + (size_t)N * FDIM * 4);        // N*128 f32 (agg1)
    unsigned* deg  = (unsigned*)(ws + (size_t)2 * N * FDIM * 4); // N u32
    float*    dinv = (float*)((char*)deg + (size_t)N * 4);       // N f32
    float*    accum = (float*)((char*)dinv + (size_t)N * 4);     // 2 f32
    unsigned* cnt   = (unsigned*)(accum + 2);                    // 2 u32

    float* rep      = (float*)d_out;            // N*128
    float* loss_out = rep + (size_t)N * FDIM;   // 1

    const int nstrips = N >> 4;
    const int gblocks = (nstrips + 7) / 8;

    // degree + dinv (shared by both layers)
    zero_init_kernel<<<(N + 255) / 256, 256, 0, stream>>>(deg, accum, cnt, N);
    count_deg_kernel<<<(E + 255) / 256, 256, 0, stream>>>(dst, deg, E);
    dinv_kernel<<<(N + 255) / 256, 256, 0, stream>>>(deg, dinv, N);

    // layer 1: h1 = features @ w1 ; agg1 = h1*dinv^2 + b1 ; scatter h1 -> agg1
    gemm_wmma_kernel<false><<<gblocks, 256, 0, stream>>>(features, w1, bufA, bufB,
                                                         dinv, b1, N);
    scatter_kernel<<<(E + 7) / 8, 256, 0, stream>>>((const float4*)bufA, bufB, src, dst, dinv, E);

    // layer 2: h2 = relu(agg1) @ w2 ; rep = h2*dinv^2 + b2 ; scatter h2 -> rep
    gemm_wmma_kernel<true><<<gblocks, 256, 0, stream>>>(bufB, w2, bufA, rep,
                                                        dinv, b2, N);
    scatter_kernel<<<(E + 7) / 8, 256, 0, stream>>>((const float4*)bufA, rep, src, dst, dinv, E);

    // loss
    edge_loss_kernel<<<2048, 256, 0, stream>>>((const float4*)rep, src, dst, E, 1.0f,
                                               &accum[0], &cnt[0]);
    edge_loss_kernel<<<2048, 256, 0, stream>>>((const float4*)rep, nsrc, ndst, En, 0.0f,
                                               &accum[1], &cnt[1]);
    finalize_kernel<<<1, 1, 0, stream>>>(accum, cnt, loss_out, (float)N);
}